// ROIPoolling_28432683499544
// MI455X (gfx1250) — compile-verified
//
#include <hip/hip_runtime.h>
#include <math.h>

// ROI max-pool for MI455X (gfx1250, wave32).
// One block (8 waves) per ROI; each wave owns channels c ≡ wave (mod 8).
// Per group of 4 channels: issue all global_load_async_to_lds_b32 staging
// (CDNA5 async data mover, ASYNCcnt) into 4 wave-private LDS sub-buffers,
// one s_wait_asynccnt 0, then reduce all 4 from LDS with wave32 shuffle
// trees. Per-lane region offsets (the only integer divisions) are computed
// ONCE per ROI and reused across all 64 channels a wave handles.
// No barriers anywhere: buffers are wave-private, waits are per-wave counters.

#define IMG_W_F 1024.0f
#define IMG_H_F 800.0f
#define NWAVES  8
#define BLOCKSZ 256
#define RBUF    128          // floats per channel sub-buffer (region cap)
#define KMAX    (RBUF / 32)  // async instrs per channel stage (4)
#define CHU     4            // channels staged per wait (MLP factor)

__global__ __launch_bounds__(BLOCKSZ) void roi_maxpool_async_kernel(
    const float* __restrict__ feat,   // [B, C, H, W]
    const float* __restrict__ rois,   // [B, N, 4] (x1,y1,x2,y2) pixel coords
    float* __restrict__ out,          // [B, N, C]
    int C, int H, int W, int N)
{
    __shared__ float smem[NWAVES * CHU * RBUF];   // 16 KB

    const int bn = blockIdx.x;        // flat (b*N + n)
    const int b  = bn / N;

    // Uniform per-block ROI decode (stays scalar).
    const float4 r = reinterpret_cast<const float4*>(rois)[bn];

    int x1 = (int)floorf(r.x / IMG_W_F * (float)W); if (x1 < 0) x1 = 0;
    int y1 = (int)floorf(r.y / IMG_H_F * (float)H); if (y1 < 0) y1 = 0;
    int x2 = (int)ceilf (r.z / IMG_W_F * (float)W); if (x2 < 0) x2 = 0;
    int y2 = (int)ceilf (r.w / IMG_H_F * (float)H); if (y2 < 0) y2 = 0;
    if (x1 == 0 && x2 == 0) x2 = 1;   // degenerate-box fixes (reference order)
    if (y1 == 0 && y2 == 0) y2 = 1;
    if (x1 >= W) x1 = W - 1;
    if (y1 >= H) y1 = H - 1;
    const int x2e = x2 < W ? x2 : W;  // half-open slice clip
    const int y2e = y2 < H ? y2 : H;
    const int rw  = x2e - x1;
    const int rh  = y2e - y1;
    const int rsize = (rw > 0 && rh > 0) ? rw * rh : 0;

    const int wave = (int)(threadIdx.x >> 5);   // wave32
    const int lane = (int)(threadIdx.x & 31);

    float* outp = out + (size_t)bn * (size_t)C;

    if (rsize == 0) {                 // empty region -> -inf (matches reference)
        for (int c = wave; c < C; c += NWAVES)
            if (lane == 0) outp[c] = -INFINITY;
        return;
    }

    const size_t planeStride = (size_t)(H * W);
    const float* plane0 = feat + (size_t)b * (size_t)C * planeStride
                               + (size_t)y1 * (size_t)W + (size_t)x1;

    if (rsize <= RBUF) {
        // ---- Per-ROI precompute: lane -> (row,col) byte offsets. The only
        //      integer divisions in the kernel; executed once, reused for all
        //      channels this wave reduces. ----
        unsigned goffB[KMAX];
        bool     valid[KMAX];
        float*   sbase_f = &smem[wave * CHU * RBUF];
        // Low 32 bits of the flat shared address == LDS byte offset.
        const unsigned sbaseB = (unsigned)(size_t)sbase_f;
        #pragma unroll
        for (int j = 0; j < KMAX; ++j) {
            const int i  = lane + 32 * j;
            const int ry = i / rw;
            const int rx = i - ry * rw;
            goffB[j] = (unsigned)((ry * W + rx) * 4);
            valid[j] = (i < rsize);
        }

        // ---- Channel groups: stage CHU channels, one wait, reduce CHU. ----
        for (int t = wave; t < C; t += NWAVES * CHU) {
            #pragma unroll
            for (int u = 0; u < CHU; ++u) {
                const int c = t + NWAVES * u;
                if (c < C) {
                    const char* srcb =
                        (const char*)(plane0 + (size_t)c * planeStride);
                    const unsigned lbase =
                        sbaseB + (unsigned)(u * RBUF * 4) + (unsigned)lane * 4u;
                    #pragma unroll
                    for (int j = 0; j < KMAX; ++j) {
                        if (valid[j]) {
                            const unsigned loff = lbase + (unsigned)(32 * j) * 4u;
                            const char* g = srcb + goffB[j];
                            asm volatile(
                                "global_load_async_to_lds_b32 %0, %1, off"
                                :: "v"(loff), "v"(g) : "memory");
                        }
                    }
                }
            }
            asm volatile("s_wait_asynccnt 0" ::: "memory");
            #pragma unroll
            for (int u = 0; u < CHU; ++u) {
                const int c = t + NWAVES * u;
                if (c < C) {
                    const float* sb = sbase_f + u * RBUF;
                    float m = -INFINITY;
                    #pragma unroll
                    for (int j = 0; j < KMAX; ++j)
                        if (valid[j]) m = fmaxf(m, sb[lane + 32 * j]);
                    #pragma unroll
                    for (int off = 16; off > 0; off >>= 1)
                        m = fmaxf(m, __shfl_xor(m, off, 32));
                    if (lane == 0) outp[c] = m;
                }
            }
        }
    } else {
        // ---- Oversized-region fallback: direct global reduction, nested
        //      row loops (division-free). ----
        for (int c = wave; c < C; c += NWAVES) {
            const float* src = plane0 + (size_t)c * planeStride;
            float m = -INFINITY;
            for (int ry = 0; ry < rh; ++ry) {
                const float* row = src + (size_t)ry * (size_t)W;
                for (int rx = lane; rx < rw; rx += 32)
                    m = fmaxf(m, row[rx]);
            }
            #pragma unroll
            for (int off = 16; off > 0; off >>= 1)
                m = fmaxf(m, __shfl_xor(m, off, 32));
            if (lane == 0) outp[c] = m;
        }
    }
}

extern "C" void kernel_launch(void* const* d_in, const int* in_sizes, int n_in,
                              void* d_out, int out_size, void* d_ws, size_t ws_size,
                              hipStream_t stream) {
    (void)n_in; (void)out_size; (void)d_ws; (void)ws_size;
    const float* feat = (const float*)d_in[0];
    const float* rois = (const float*)d_in[1];
    float* out = (float*)d_out;

    const int C = 512, H = 50, W = 64;
    int B = in_sizes[0] / (C * H * W);
    if (B < 1) B = 1;
    int N = in_sizes[1] / (4 * B);
    if (N < 1) N = 1;

    dim3 grid((unsigned)(B * N));
    dim3 block(BLOCKSZ);
    hipLaunchKernelGGL(roi_maxpool_async_kernel, grid, block, 0, stream,
                       feat, rois, out, C, H, W, N);
}